// Attention_69965017252614
// MI455X (gfx1250) — compile-verified
//
#include <hip/hip_runtime.h>
#include <hip/hip_bf16.h>
#include <math.h>

#define H_ 8
#define L_ 2048
#define D_ 64
#define NKT (L_/16)   // 128 k-tiles

typedef float v2f __attribute__((ext_vector_type(2)));
typedef float v8f __attribute__((ext_vector_type(8)));

union V8 { v8f v; float f[8]; };

__device__ __forceinline__ v8f wmma4(v2f a, v2f b, v8f c) {
  // D = A(16x4,f32) * B(4x16,f32) + C(16x16,f32)
  return __builtin_amdgcn_wmma_f32_16x16x4_f32(false, a, false, b, (short)0, c,
                                               false, false);
}

// Branch-free, numerically stable softplus: max(x,0) + log1p(exp(-|x|)).
// No EXEC-mask manipulation -> safe in the same loop body as WMMA chains.
__device__ __forceinline__ float softplus_f(float x) {
  return fmaxf(x, 0.0f) + __logf(1.0f + __expf(-fabsf(x)));
}

__global__ __launch_bounds__(128) void attn_bias_softmax_kernel(
    const float* __restrict__ Q, const float* __restrict__ K,
    const float* __restrict__ V, const float* __restrict__ Cw,
    const float* __restrict__ DQ, const float* __restrict__ DKT,
    const float* __restrict__ DKB, const float* __restrict__ DKS,
    const float* __restrict__ RT, const float* __restrict__ RB,
    const float* __restrict__ WW, const float* __restrict__ WB,
    float* __restrict__ out, float* __restrict__ probs)
{
  __shared__ float s_stat[4 * 16];        // per-wave row max, then row sums
  __shared__ float s_tile[4][256];        // per-wave 16x16 ec tile (LDS transpose)
  __shared__ float s_osum[4][16 * 64];    // per-wave partial O

  const int tid  = threadIdx.x;
  const int wave = tid >> 5;
  const int lane = tid & 31;
  const int half = lane >> 4;   // 0: lanes 0-15, 1: lanes 16-31
  const int lk   = lane & 15;   // column (key) index inside a 16-wide tile

  const int h  = blockIdx.x >> 7;         // 128 q-tiles per head
  const int q0 = (blockIdx.x & 127) << 4; // first query row of this tile

  const float scale = 0.125f;             // 1/sqrt(64)

  // ---- preload Q rows in WMMA A-operand layout (16x4 chunks over D=64) ----
  v2f aq[16];
  {
    const float* qrow = Q + ((size_t)(h * L_) + q0 + lk) * D_;
    #pragma unroll
    for (int dd = 0; dd < 16; ++dd) {
      int base = dd * 4 + 2 * half;
      aq[dd].x = qrow[base];
      aq[dd].y = qrow[base + 1];
    }
  }

  // ---- preload d_q for this half's 8 query rows ----
  float dqv[8][3];
  #pragma unroll
  for (int r = 0; r < 8; ++r) {
    int qq = q0 + r + 8 * half;
    dqv[r][0] = DQ[qq * 3 + 0];
    dqv[r][1] = DQ[qq * 3 + 1];
    dqv[r][2] = DQ[qq * 3 + 2];
  }

  const float ww0 = WW[0 * H_ + h], ww1 = WW[1 * H_ + h];
  const float ww2 = WW[2 * H_ + h], ww3 = WW[3 * H_ + h];
  const float wb0 = WB[0 * H_ + h], wb1 = WB[1 * H_ + h];
  const float wb2 = WB[2 * H_ + h], wb3 = WB[3 * H_ + h];

  float rowmax[8];
  #pragma unroll
  for (int r = 0; r < 8; ++r) rowmax[r] = -1e30f;

  // ================= pass 1: biased scores + row max =================
  for (int kt = wave; kt < NKT; kt += 4) {
    const int k0   = kt << 4;
    const int kcol = k0 + lk;

    // S = Q * K^T  (accumulate over D in 16 chunks of 4)
    V8 acc;
    #pragma unroll
    for (int i = 0; i < 8; ++i) acc.f[i] = 0.0f;
    {
      const float* krow = K + ((size_t)(h * L_) + kcol) * D_;
      #pragma unroll
      for (int dd = 0; dd < 16; ++dd) {
        int base = dd * 4 + 2 * half;
        v2f bb;
        bb.x = krow[base];
        bb.y = krow[base + 1];
        acc.v = wmma4(aq[dd], bb, acc.v);
      }
    }

    // per-key parameters (depend only on kcol -> one set per lane)
    const float* ks = DKS + (size_t)kcol * 8;
    const float st0 = ks[0], st1 = ks[1], st2 = ks[2], st3 = ks[3];
    const float sb0 = ks[4], sb1 = ks[5], sb2 = ks[6], sb3 = ks[7];
    const float* pkt = DKT + (size_t)kcol * 3;
    const float* pkb = DKB + (size_t)kcol * 3;
    const float kt0 = pkt[0], kt1 = pkt[1], kt2 = pkt[2];
    const float kb0 = pkb[0], kb1 = pkb[1], kb2 = pkb[2];

    #pragma unroll
    for (int r = 0; r < 8; ++r) {
      const int qq = q0 + r + 8 * half;
      const float dm0 = (dqv[r][0] - kt0) * st0 + (dqv[r][0] - kb0) * sb0;
      const float dm1 = (dqv[r][1] - kt1) * st1 + (dqv[r][1] - kb1) * sb1;
      const float dm2 = (dqv[r][2] - kt2) * st2 + (dqv[r][2] - kb2) * sb2;
      const size_t ridx = (size_t)qq * L_ + kcol;
      const float dm3 = RT[ridx] * st3 + RB[ridx] * sb3;
      const float wl = dm0 * ww0 + dm1 * ww1 + dm2 * ww2 + dm3 * ww3;
      const float bl = dm0 * wb0 + dm1 * wb1 + dm2 * wb2 + dm3 * wb3;
      const float sc = acc.f[r] * scale * softplus_f(wl) + bl;
      probs[((size_t)(h * L_) + qq) * L_ + kcol] = sc;   // scratch: raw score
      rowmax[r] = fmaxf(rowmax[r], sc);
    }
  }

  // reduce row max across the 16 lanes of each half, then across waves
  #pragma unroll
  for (int m = 1; m < 16; m <<= 1) {
    #pragma unroll
    for (int r = 0; r < 8; ++r)
      rowmax[r] = fmaxf(rowmax[r], __shfl_xor(rowmax[r], m, 32));
  }
  if (lk == 0) {
    #pragma unroll
    for (int r = 0; r < 8; ++r) s_stat[wave * 16 + r + 8 * half] = rowmax[r];
  }
  __syncthreads();
  float mrow[8];
  #pragma unroll
  for (int r = 0; r < 8; ++r) {
    const int qq = r + 8 * half;
    float mv = s_stat[qq];
    mv = fmaxf(mv, s_stat[16 + qq]);
    mv = fmaxf(mv, s_stat[32 + qq]);
    mv = fmaxf(mv, s_stat[48 + qq]);
    mrow[r] = mv;
  }
  __syncthreads();   // s_stat reused for sums below

  // ================= pass 2: exp, gate by c, O += ec * V =================
  float rowsum[8];
  #pragma unroll
  for (int r = 0; r < 8; ++r) rowsum[r] = 0.0f;
  V8 o[4];
  #pragma unroll
  for (int dt = 0; dt < 4; ++dt)
    #pragma unroll
    for (int i = 0; i < 8; ++i) o[dt].f[i] = 0.0f;

  float* tile = s_tile[wave];
  for (int kt = wave; kt < NKT; kt += 4) {
    const int k0   = kt << 4;
    const int kcol = k0 + lk;
    const float cr = Cw[kcol];

    #pragma unroll
    for (int r = 0; r < 8; ++r) {
      const int qq = q0 + r + 8 * half;
      const size_t pidx = ((size_t)(h * L_) + qq) * L_ + kcol;
      const float e = __expf(probs[pidx] - mrow[r]);
      rowsum[r] += e;
      const float ec = e * cr;
      probs[pidx] = ec;                        // scratch: gated unnormalized
      tile[(r + 8 * half) * 16 + lk] = ec;     // C-layout -> LDS
    }
    asm volatile("s_wait_dscnt 0" ::: "memory");

    // read back in WMMA A-operand layout: A = ec[:, kc*4 .. kc*4+3]
    v2f ap[4];
    #pragma unroll
    for (int kc = 0; kc < 4; ++kc) {
      const int kb = kc * 4 + 2 * half;
      ap[kc].x = tile[lk * 16 + kb];
      ap[kc].y = tile[lk * 16 + kb + 1];
    }
    asm volatile("s_wait_dscnt 0" ::: "memory");

    #pragma unroll
    for (int dt = 0; dt < 4; ++dt) {
      #pragma unroll
      for (int kc = 0; kc < 4; ++kc) {
        const float* vp = V + ((size_t)(h * L_) + k0 + kc * 4 + 2 * half) * D_
                            + dt * 16 + lk;
        v2f bb;
        bb.x = vp[0];
        bb.y = vp[D_];
        o[dt].v = wmma4(ap[kc], bb, o[dt].v);
      }
    }
  }

  // reduce row sums across lanes, publish per-wave sums + partial O
  #pragma unroll
  for (int m = 1; m < 16; m <<= 1) {
    #pragma unroll
    for (int r = 0; r < 8; ++r)
      rowsum[r] += __shfl_xor(rowsum[r], m, 32);
  }
  if (lk == 0) {
    #pragma unroll
    for (int r = 0; r < 8; ++r) s_stat[wave * 16 + r + 8 * half] = rowsum[r];
  }
  #pragma unroll
  for (int dt = 0; dt < 4; ++dt)
    #pragma unroll
    for (int r = 0; r < 8; ++r)
      s_osum[wave][(r + 8 * half) * 64 + dt * 16 + lk] = o[dt].f[r];
  __syncthreads();

  // ---- final out = (sum_w O_w) / rowsum_total ----
  for (int i = tid; i < 16 * 64; i += 128) {
    const int qq = i >> 6;
    const int dd = i & 63;
    const float s = s_osum[0][i] + s_osum[1][i] + s_osum[2][i] + s_osum[3][i];
    const float tot = s_stat[qq] + s_stat[16 + qq] + s_stat[32 + qq] + s_stat[48 + qq];
    out[((size_t)(h * L_) + q0 + qq) * D_ + dd] = s / tot;
  }

  // ================= pass 3: normalize probs =================
  float invt[8];
  #pragma unroll
  for (int r = 0; r < 8; ++r) {
    const int qq = r + 8 * half;
    invt[r] = 1.0f / (s_stat[qq] + s_stat[16 + qq] + s_stat[32 + qq] + s_stat[48 + qq]);
  }
  for (int kt = wave; kt < NKT; kt += 4) {
    const int kcol = (kt << 4) + lk;
    #pragma unroll
    for (int r = 0; r < 8; ++r) {
      const int qq = q0 + r + 8 * half;
      const size_t pidx = ((size_t)(h * L_) + qq) * L_ + kcol;
      probs[pidx] *= invt[r];
    }
  }
}

extern "C" void kernel_launch(void* const* d_in, const int* in_sizes, int n_in,
                              void* d_out, int out_size, void* d_ws, size_t ws_size,
                              hipStream_t stream) {
  const float* Q   = (const float*)d_in[0];
  const float* K   = (const float*)d_in[1];
  const float* V   = (const float*)d_in[2];
  const float* Cw  = (const float*)d_in[3];
  const float* DQ  = (const float*)d_in[4];
  const float* DKT = (const float*)d_in[5];
  const float* DKB = (const float*)d_in[6];
  const float* DKS = (const float*)d_in[7];
  const float* RT  = (const float*)d_in[8];
  const float* RB  = (const float*)d_in[9];
  const float* WW  = (const float*)d_in[10];
  const float* WB  = (const float*)d_in[11];

  float* out   = (float*)d_out;
  float* probs = out + (size_t)H_ * L_ * D_;   // outputs concatenated: out, probs

  dim3 grid(H_ * (L_ / 16));
  dim3 block(128);
  hipLaunchKernelGGL(attn_bias_softmax_kernel, grid, block, 0, stream,
                     Q, K, V, Cw, DQ, DKT, DKB, DKS, RT, RB, WW, WB, out, probs);
}